// OutageHawkes_31009663877513
// MI455X (gfx1250) — compile-verified
//
#include <hip/hip_runtime.h>

// OutageHawkes ground_lam: lam[b] = max(mu[xs_b] + beta * sum_j alpha[hs_bj, xs_b]
//                                        * exp(-beta*|xt_b - ht_bj|) * (ht_bj != -1), 1e-5)
// B=16384, H=1024, S=200. Memory-bound: streams h (134 MB) once.

#define B_SIZE 16384
#define H_SIZE 1024
#define S_SIZE 200
#define WAVES_PER_BLOCK 8   // 256 threads = 8 wave32; one wave per batch row

typedef __attribute__((ext_vector_type(2))) float v2f;
typedef __attribute__((ext_vector_type(8))) float v8f;

__global__ __launch_bounds__(256) void OutageHawkes_lam_kernel(
    const float* __restrict__ x,      // [B,2]  (t, s)
    const float* __restrict__ h,      // [B,H,2] (t, s)
    const float* __restrict__ alpha,  // [S,S]
    const float* __restrict__ beta_p, // [1]
    const float* __restrict__ mu,     // [S]
    float* __restrict__ lam)          // [B]
{
    const int wave = (int)(threadIdx.x >> 5);
    const int lane = (int)(threadIdx.x & 31);
    const int b    = (int)blockIdx.x * WAVES_PER_BLOCK + wave;

    const float beta = beta_p[0];
    const float nb   = -beta;
    const float xt   = x[2 * b];
    const int   xs   = (int)x[2 * b + 1];

    // alpha column for this row's mark: alpha[hs*S + xs]
    const float* __restrict__ acol = alpha + xs;
    const float4* __restrict__ hrow =
        (const float4*)(h + (size_t)b * (size_t)(2 * H_SIZE));

    float acc = 0.0f;

    // 16 coalesced b128 loads per lane; 2 events per load; EXEC stays all-1s
    // (mask is arithmetic, no divergence -> WMMA-legal at the end).
#pragma unroll
    for (int k = 0; k < H_SIZE / 64; ++k) {
        float4 v = hrow[k * 32 + lane];
        // event 0
        {
            float t  = v.x;
            int   hs = (int)v.y;                  // valid index even when padded
            float a  = acol[hs * S_SIZE];         // L2-resident gather (160 KB table)
            float m  = (t != -1.0f) ? 1.0f : 0.0f;
            float e  = __expf(nb * fabsf(xt - t)); // v_exp_f32 path
            acc = fmaf(m * a, e, acc);
        }
        // event 1
        {
            float t  = v.z;
            int   hs = (int)v.w;
            float a  = acol[hs * S_SIZE];
            float m  = (t != -1.0f) ? 1.0f : 0.0f;
            float e  = __expf(nb * fabsf(xt - t));
            acc = fmaf(m * a, e, acc);
        }
    }

    // ---- wave32 all-reduce of acc via V_WMMA_F32_16X16X4_F32 ----
    // A(16x4): VGPR0 = acc for every lane, VGPR1 = 0
    //   => A[m,0]=acc(m), A[m,2]=acc(m+16), A[m,1]=A[m,3]=0.
    // B(4x16) = ones  => D[m,n] = acc(m) + acc(m+16) for all n.
    // Lane n holds D[0..7,n], lane 16+n holds D[8..15,n]; sum own 8 + xor-16.
    float total;
#if __has_builtin(__builtin_amdgcn_wmma_f32_16x16x4_f32)
    {
        v2f a;  a[0] = acc;  a[1] = 0.0f;
        v2f bo; bo[0] = 1.0f; bo[1] = 1.0f;
        v8f c = {};
        c = __builtin_amdgcn_wmma_f32_16x16x4_f32(
                /*neg_a=*/false, a, /*neg_b=*/false, bo,
                /*c_mod=*/(short)0, c, /*reuse_a=*/false, /*reuse_b=*/false);
        float s8 = ((c[0] + c[1]) + (c[2] + c[3])) + ((c[4] + c[5]) + (c[6] + c[7]));
        total = s8 + __shfl_xor(s8, 16, 32);
    }
#else
    {
        float s = acc;
        s += __shfl_xor(s, 1, 32);
        s += __shfl_xor(s, 2, 32);
        s += __shfl_xor(s, 4, 32);
        s += __shfl_xor(s, 8, 32);
        s += __shfl_xor(s, 16, 32);
        total = s;
    }
#endif

    if (lane == 0) {
        float l = fmaf(beta, total, mu[xs]);   // beta factored out of the sum
        lam[b] = fmaxf(l, 1e-5f);
    }
}

extern "C" void kernel_launch(void* const* d_in, const int* in_sizes, int n_in,
                              void* d_out, int out_size, void* d_ws, size_t ws_size,
                              hipStream_t stream) {
    (void)in_sizes; (void)n_in; (void)d_ws; (void)ws_size; (void)out_size;
    const float* x     = (const float*)d_in[0];  // [B,2]
    const float* h     = (const float*)d_in[1];  // [B,H,2]
    const float* alpha = (const float*)d_in[2];  // [S,S]
    const float* beta  = (const float*)d_in[3];  // [1]
    const float* mu    = (const float*)d_in[4];  // [S]
    float* lam = (float*)d_out;                  // [B]

    dim3 grid(B_SIZE / WAVES_PER_BLOCK);         // 2048 blocks
    dim3 block(32 * WAVES_PER_BLOCK);            // 256 threads = 8 waves
    OutageHawkes_lam_kernel<<<grid, block, 0, stream>>>(x, h, alpha, beta, mu, lam);
}